// MultiRegionEmbeddingLayer_48885317763664
// MI455X (gfx1250) — compile-verified
//
#include <hip/hip_runtime.h>

// MultiRegionEmbeddingLayer for MI455X (gfx1250).
// Memory-bound gather + windowed max; staged via Tensor Data Mover into LDS.

#define B_    16
#define S_    1024
#define E_    128
#define R0_   7
#define RAD_  3
#define T_    64                    // tokens per block
#define ROWS_ (T_ + 2 * RAD_)       // 70 LDS rows
#define TPB_  256                   // 8 waves (wave32)

struct __align__(16) F4 { float x, y, z, w; };

__device__ __forceinline__ F4 f4mul(F4 a, F4 b) {
  F4 r; r.x = a.x * b.x; r.y = a.y * b.y; r.z = a.z * b.z; r.w = a.w * b.w; return r;
}
__device__ __forceinline__ F4 f4max(F4 a, F4 b) {
  F4 r; r.x = fmaxf(a.x, b.x); r.y = fmaxf(a.y, b.y);
  r.z = fmaxf(a.z, b.z); r.w = fmaxf(a.w, b.w); return r;
}

typedef unsigned int u32x4 __attribute__((ext_vector_type(4)));
typedef int          i32x8 __attribute__((ext_vector_type(8)));
typedef int          i32x4 __attribute__((ext_vector_type(4)));

__global__ __launch_bounds__(TPB_) void mre_kernel(
    const int*   __restrict__ seq,   // [B,S]
    const float* __restrict__ emb,   // [B,S,E]
    const float* __restrict__ U,     // [VOCAB,7,E]
    float*       __restrict__ out)   // [B,S,3E]
{
  __shared__ float lds_emb[ROWS_ * E_];   // 35840 B

  const int tid = threadIdx.x;
  const int blk = blockIdx.x;
  const int b   = blk >> 4;               // S_/T_ == 16 tiles
  const int s0  = (blk & 15) * T_;

  // Valid global row range for this tile (handles zero padding at edges).
  const int rstart = (s0 - RAD_ < 0) ? 0 : (s0 - RAD_);
  const int rend   = (s0 + T_ + RAD_ > S_) ? S_ : (s0 + T_ + RAD_);
  const int nrows  = rend - rstart;       // 67 or 70
  const int joff   = rstart - (s0 - RAD_);// leading pad rows in LDS (0 or 3)
  const int padrows = ROWS_ - nrows;      // 0 or 3

  // Zero the pad rows (only first/last tile of each batch row has any).
  for (int c = tid; c < padrows * (E_ / 4); c += TPB_) {
    int pr  = c >> 5;                     // pad row 0..padrows-1
    int j   = (joff > 0) ? pr : (nrows + pr);
    int col = c & 31;
    F4 z = {0.f, 0.f, 0.f, 0.f};
    *(F4*)&lds_emb[j * E_ + col * 4] = z;
  }

#if __has_builtin(__builtin_amdgcn_tensor_load_to_lds)
  // One TDM op stages the whole [nrows x 128] f32 tile into LDS.
  if (tid < 32) {
    // Generic LDS pointer: low 32 bits are the workgroup-relative LDS offset.
    unsigned lds_addr = (unsigned)(unsigned long long)(size_t)&lds_emb[joff * E_];
    unsigned long long ga =
        (unsigned long long)(size_t)(emb + ((size_t)b * S_ + rstart) * E_);

    // D# group 0: count=1 | lds_addr | global_addr[56:0] | type=2 ("image")
    u32x4 g0 = { 1u,
                 lds_addr,
                 (unsigned)(ga & 0xffffffffu),
                 (unsigned)((ga >> 32) & 0x01ffffffu) | (2u << 30) };

    // D# group 1: data_size=2 (4B); tensor 128 x nrows; tile 128 x nrows;
    //             tensor_dim0_stride = 128 elements.
    i32x8 g1 = { (int)(2u << 16),                              // wg_mask=0, data_size=4B
                 (int)((unsigned)E_ << 16),                    // tensor_dim0 lo16 (=128)
                 (int)(((unsigned)nrows & 0xffffu) << 16),     // dim0 hi16=0 | tensor_dim1 lo16
                 (int)((unsigned)E_ << 16),                    // dim1 hi16=0 | tile_dim0=128
                 (int)((unsigned)nrows & 0xffffu),             // tile_dim1=nrows, tile_dim2=0
                 (int)E_,                                      // tensor_dim0_stride lo32 = 128
                 0,                                            // stride0 hi16 | stride1 lo16
                 0 };                                          // stride1 hi32 (unused, 2D)

    i32x4 z4 = { 0, 0, 0, 0 };                                 // groups 2/3 unused (2D)
    i32x8 z8 = { 0, 0, 0, 0, 0, 0, 0, 0 };
    __builtin_amdgcn_tensor_load_to_lds(g0, g1, z4, z4, z8, 0);

#if __has_builtin(__builtin_amdgcn_s_wait_tensorcnt)
    __builtin_amdgcn_s_wait_tensorcnt(0);
#else
    asm volatile("s_wait_tensorcnt 0" ::: "memory");
#endif
  }
#else
  // Fallback: vectorized cooperative copy.
  for (int c = tid; c < nrows * (E_ / 4); c += TPB_) {
    int j   = c >> 5;
    int col = c & 31;
    *(F4*)&lds_emb[(joff + j) * E_ + col * 4] =
        *(const F4*)(emb + ((size_t)b * S_ + rstart + j) * E_ + col * 4);
  }
#endif

  __syncthreads();

  // One wave per token; one lane per 4 output channels.
  const int wave = tid >> 5;
  const int lane = tid & 31;
  const int e4   = lane << 2;

  for (int tt = wave; tt < T_; tt += (TPB_ / 32)) {
    const int s    = s0 + tt;
    const int vidx = seq[b * S_ + s];
    const float* Urow = U + (size_t)vidx * (R0_ * E_) + e4;

    // Prefetch next token's U rows into cache (global_prefetch_b8).
    if (tt + (TPB_ / 32) < T_) {
      int nidx = seq[b * S_ + s + (TPB_ / 32)];
      const float* Un = U + (size_t)nidx * (R0_ * E_) + e4;
#pragma unroll
      for (int i = 0; i < R0_; ++i)
        __builtin_prefetch(Un + i * E_, 0, 0);
    }

    F4 p[R0_];
#pragma unroll
    for (int i = 0; i < R0_; ++i) {
      F4 u = *(const F4*)(Urow + i * E_);                     // global_load_b128 (L2-hot)
      F4 x = *(const F4*)&lds_emb[(tt + i) * E_ + e4];        // ds_load_b128
      p[i] = f4mul(u, x);
    }

    F4 m3 = f4max(p[2], f4max(p[3], p[4]));   // window 3 (i=2..4)
    F4 m5 = f4max(m3, f4max(p[1], p[5]));     // window 5 (i=1..5)
    F4 m7 = f4max(m5, f4max(p[0], p[6]));     // window 7 (i=0..6)

    float* ob = out + ((size_t)b * S_ + s) * (3 * E_) + e4;
    *(F4*)(ob)          = m7;                  // r=7 slice first
    *(F4*)(ob + E_)     = m5;                  // then r=5
    *(F4*)(ob + 2 * E_) = m3;                  // then r=3
  }
}

extern "C" void kernel_launch(void* const* d_in, const int* in_sizes, int n_in,
                              void* d_out, int out_size, void* d_ws, size_t ws_size,
                              hipStream_t stream) {
  (void)in_sizes; (void)n_in; (void)out_size; (void)d_ws; (void)ws_size;
  const int*   seq = (const int*)d_in[0];
  const float* emb = (const float*)d_in[1];
  const float* U   = (const float*)d_in[2];
  float*       out = (float*)d_out;

  dim3 grid(B_ * (S_ / T_));   // 256 blocks
  mre_kernel<<<grid, TPB_, 0, stream>>>(seq, emb, U, out);
}